// RMultiHeadGraphAttention3_52716428591540
// MI455X (gfx1250) — compile-verified
//
#include <hip/hip_runtime.h>
#include <math.h>

#define N_NODES 4096
#define N_EDGES 8192
#define FDIM    512
#define N_HEAD  4

typedef __attribute__((ext_vector_type(2))) float v2f;
typedef __attribute__((ext_vector_type(8))) float v8f;

// ---------------------------------------------------------------------------
// K1: inputr[rel] += (h[src]-h[dst])/r_count[e]   (scatter over relation ids)
// ---------------------------------------------------------------------------
__global__ void scatter_edge_h_kernel(const float* __restrict__ h,
                                      const int*   __restrict__ A,
                                      const float* __restrict__ r_count,
                                      float*       __restrict__ inputr) {
  int e   = blockIdx.x;
  int dst = A[e];
  int rel = A[N_EDGES + e];
  int src = A[2 * N_EDGES + e];
  float inv = 1.0f / r_count[e];
  const float* hs = h + (size_t)src * FDIM;
  const float* hd = h + (size_t)dst * FDIM;
  float* row = inputr + (size_t)rel * FDIM;
  for (int f = threadIdx.x; f < FDIM; f += blockDim.x)
    atomicAdd(&row[f], (hs[f] - hd[f]) * inv);
}

// ---------------------------------------------------------------------------
// K2: edge_r = inputr / r_count[j]  (in place)  and  s[f] += edge_r[j,f]
// ---------------------------------------------------------------------------
__global__ void finalize_edge_r_kernel(float* __restrict__ edge_r,
                                       const float* __restrict__ r_count,
                                       float* __restrict__ svec) {
  int j = blockIdx.x;
  float inv = 1.0f / r_count[j];
  float* row = edge_r + (size_t)j * FDIM;
  for (int f = threadIdx.x; f < FDIM; f += blockDim.x) {
    float v = row[f] * inv;
    row[f] = v;
    atomicAdd(&svec[f], v);
  }
}

// ---------------------------------------------------------------------------
// K3: rowsum[j] = edge_r[j,:] . s ;  sc[j] = r_count[j]/rowsum[j]
// (aij.sum(1)[j] = edge_r[j] . sum_k edge_r[k] -- no ExE matrix needed)
// ---------------------------------------------------------------------------
__global__ void rowscale_kernel(const float* __restrict__ edge_r,
                                const float* __restrict__ svec,
                                const float* __restrict__ r_count,
                                float* __restrict__ sc) {
  __shared__ float red[256];
  int j = blockIdx.x;
  const float* row = edge_r + (size_t)j * FDIM;
  float p = 0.0f;
  for (int f = threadIdx.x; f < FDIM; f += 256) p += row[f] * svec[f];
  red[threadIdx.x] = p;
  __syncthreads();
  for (int s2 = 128; s2 > 0; s2 >>= 1) {
    if (threadIdx.x < s2) red[threadIdx.x] += red[threadIdx.x + s2];
    __syncthreads();
  }
  if (threadIdx.x == 0) sc[j] = r_count[j] / red[0];
}

// ---------------------------------------------------------------------------
// K4 (WMMA): T[m,n] = sum_e (sc[e]*edge_r[e,m]) * edge_r[e,n]     [512x512]
// Each wave32 owns a 16x64 strip: 4 accumulators share one A fragment.
// 32 M-tiles * 8 N-strips = 256 waves = 32 blocks * 8 waves.
// V_WMMA_F32_16X16X4_F32 fragment layout (wave32):
//   A 16x4: lanes 0-15 -> M=lane, K={0,1}; lanes 16-31 -> M=lane-16, K={2,3}
//   B 4x16: same K striping, N=lane&15
//   C/D:    VGPR v, lanes 0-15 -> M=v, lanes 16-31 -> M=v+8
// ---------------------------------------------------------------------------
__global__ void gemm_syrk_wmma(const float* __restrict__ edge_r,
                               const float* __restrict__ sc,
                               float* __restrict__ T) {
  int wave  = threadIdx.x >> 5;
  int lane  = threadIdx.x & 31;
  int strip = blockIdx.x * 8 + wave;      // 0..255
  int tm = (strip >> 3) << 4;             // M base: 0..496
  int tn = (strip & 7) << 6;              // N base: 0..448 (64-wide strip)
  int half = lane >> 4;                   // 0 | 1
  int l16  = lane & 15;
  int kb   = half * 2;                    // K offset within group-of-4
  int mg   = tm + l16;

  v8f acc0 = {}, acc1 = {}, acc2 = {}, acc3 = {};
#pragma unroll 2
  for (int k0 = 0; k0 < N_EDGES; k0 += 4) {
    int r0 = k0 + kb, r1 = r0 + 1;
    // fold diagonal sc into the A side (2 mults/step instead of 8)
    v2f a;
    a.x = edge_r[(size_t)r0 * FDIM + mg] * sc[r0];
    a.y = edge_r[(size_t)r1 * FDIM + mg] * sc[r1];
    const float* p0 = edge_r + (size_t)r0 * FDIM + tn + l16;
    const float* p1 = edge_r + (size_t)r1 * FDIM + tn + l16;
    v2f b0, b1, b2, b3;
    b0.x = p0[0];  b0.y = p1[0];
    b1.x = p0[16]; b1.y = p1[16];
    b2.x = p0[32]; b2.y = p1[32];
    b3.x = p0[48]; b3.y = p1[48];
    acc0 = __builtin_amdgcn_wmma_f32_16x16x4_f32(false, a, false, b0, (short)0, acc0, false, false);
    acc1 = __builtin_amdgcn_wmma_f32_16x16x4_f32(false, a, false, b1, (short)0, acc1, false, false);
    acc2 = __builtin_amdgcn_wmma_f32_16x16x4_f32(false, a, false, b2, (short)0, acc2, false, false);
    acc3 = __builtin_amdgcn_wmma_f32_16x16x4_f32(false, a, false, b3, (short)0, acc3, false, false);
  }
  int row0 = tm + half * 8;
  float* o = T + (size_t)row0 * FDIM + tn + l16;
#pragma unroll
  for (int v = 0; v < 8; ++v) {
    o[(size_t)v * FDIM + 0]  = acc0[v];
    o[(size_t)v * FDIM + 16] = acc1[v];
    o[(size_t)v * FDIM + 32] = acc2[v];
    o[(size_t)v * FDIM + 48] = acc3[v];
  }
}

// ---------------------------------------------------------------------------
// K5 (WMMA): inputr_new = edge_r @ T        [8192x512] @ [512x512]
// 16x64 strip per wave: 512 M-tiles * 8 N-strips = 4096 waves = 512 blocks.
// A fragment is K-contiguous -> compiler merges into global_load_b64.
// ---------------------------------------------------------------------------
__global__ void gemm_inew_wmma(const float* __restrict__ edge_r,
                               const float* __restrict__ T,
                               float* __restrict__ inew) {
  int wave  = threadIdx.x >> 5;
  int lane  = threadIdx.x & 31;
  int strip = blockIdx.x * 8 + wave;      // 0..4095
  int tm = (strip >> 3) << 4;             // edge-row base: 0..8176
  int tn = (strip & 7) << 6;              // feature base: 0..448
  int half = lane >> 4;
  int l16  = lane & 15;
  int kb   = half * 2;
  int mrow = tm + l16;

  v8f acc0 = {}, acc1 = {}, acc2 = {}, acc3 = {};
#pragma unroll 2
  for (int k0 = 0; k0 < FDIM; k0 += 4) {
    int ka = k0 + kb;
    v2f a;
    a.x = edge_r[(size_t)mrow * FDIM + ka];
    a.y = edge_r[(size_t)mrow * FDIM + ka + 1];
    const float* q0 = T + (size_t)ka * FDIM + tn + l16;
    const float* q1 = q0 + FDIM;
    v2f b0, b1, b2, b3;
    b0.x = q0[0];  b0.y = q1[0];
    b1.x = q0[16]; b1.y = q1[16];
    b2.x = q0[32]; b2.y = q1[32];
    b3.x = q0[48]; b3.y = q1[48];
    acc0 = __builtin_amdgcn_wmma_f32_16x16x4_f32(false, a, false, b0, (short)0, acc0, false, false);
    acc1 = __builtin_amdgcn_wmma_f32_16x16x4_f32(false, a, false, b1, (short)0, acc1, false, false);
    acc2 = __builtin_amdgcn_wmma_f32_16x16x4_f32(false, a, false, b2, (short)0, acc2, false, false);
    acc3 = __builtin_amdgcn_wmma_f32_16x16x4_f32(false, a, false, b3, (short)0, acc3, false, false);
  }
  int row0 = tm + half * 8;
  float* o = inew + (size_t)row0 * FDIM + tn + l16;
#pragma unroll
  for (int v = 0; v < 8; ++v) {
    o[(size_t)v * FDIM + 0]  = acc0[v];
    o[(size_t)v * FDIM + 16] = acc1[v];
    o[(size_t)v * FDIM + 32] = acc2[v];
    o[(size_t)v * FDIM + 48] = acc3[v];
  }
}

// ---------------------------------------------------------------------------
// K6: cumulative diagonal head weights.
// cw[0]=1, cw[i]=prod_{k<i} w[k];  asrc[i]=cw[i]*a_src_dst[i,0,:]
// ---------------------------------------------------------------------------
__global__ void prep_head_kernel(const float* __restrict__ w,       // [3,1,F]
                                 const float* __restrict__ a_sd,    // [4,2,F,1]
                                 float* __restrict__ cw,            // [4,F]
                                 float* __restrict__ asrc) {        // [4,F]
  int f = blockIdx.x * blockDim.x + threadIdx.x;
  if (f >= FDIM) return;
  float c = 1.0f;
#pragma unroll
  for (int i = 0; i < N_HEAD; ++i) {
    cw[i * FDIM + f]   = c;
    asrc[i * FDIM + f] = c * a_sd[(i * 2 + 0) * FDIM + f];
    if (i < N_HEAD - 1) c *= w[i * FDIM + f];
  }
}

// ---------------------------------------------------------------------------
// K7: per-(head,edge): score -> edge_e -> scatter e_rowsum + h_prime (into out)
// ---------------------------------------------------------------------------
__global__ void head_edge_kernel(const float* __restrict__ h,
                                 const int*   __restrict__ A,
                                 const float* __restrict__ inew,
                                 const float* __restrict__ cw,
                                 const float* __restrict__ asrc,
                                 const float* __restrict__ a_sd,
                                 float* __restrict__ e_rowsum,      // [4,N]
                                 float* __restrict__ out) {         // [4,N,F]
  __shared__ float red[256];
  int e    = blockIdx.x;
  int head = blockIdx.y;
  int dst = A[e];
  int rel = A[N_EDGES + e];
  int src = A[2 * N_EDGES + e];
  const float* hs  = h    + (size_t)src * FDIM;
  const float* ir  = inew + (size_t)rel * FDIM;
  const float* as  = asrc + (size_t)head * FDIM;
  const float* ad  = a_sd + (size_t)(head * 2 + 1) * FDIM;
  const float* cwh = cw   + (size_t)head * FDIM;

  float p = 0.0f;
  for (int f = threadIdx.x; f < FDIM; f += 256)
    p += hs[f] * as[f] + ir[f] * ad[f];
  red[threadIdx.x] = p;
  __syncthreads();
  for (int s2 = 128; s2 > 0; s2 >>= 1) {
    if (threadIdx.x < s2) red[threadIdx.x] += red[threadIdx.x + s2];
    __syncthreads();
  }
  float score = red[0];
  float lr = score >= 0.0f ? score : 0.2f * score;    // leaky_relu(0.2)
  float ee = __expf(-lr);
  if (threadIdx.x == 0) atomicAdd(&e_rowsum[head * N_NODES + dst], ee);

  float* orow = out + ((size_t)head * N_NODES + dst) * FDIM;
  for (int f = threadIdx.x; f < FDIM; f += 256)
    atomicAdd(&orow[f], (hs[f] * cwh[f] - ir[f]) * ee);
}

// ---------------------------------------------------------------------------
// K8: out[i,n,f] /= e_rowsum[i,n]   (in place)
// ---------------------------------------------------------------------------
__global__ void normalize_kernel(float* __restrict__ out,
                                 const float* __restrict__ e_rowsum) {
  size_t i = (size_t)blockIdx.x * blockDim.x + threadIdx.x;
  size_t total = (size_t)N_HEAD * N_NODES * FDIM;
  if (i >= total) return;
  out[i] /= e_rowsum[i / FDIM];
}

// ---------------------------------------------------------------------------
extern "C" void kernel_launch(void* const* d_in, const int* in_sizes, int n_in,
                              void* d_out, int out_size, void* d_ws, size_t ws_size,
                              hipStream_t stream) {
  const float* h       = (const float*)d_in[0];   // [4096,512]
  const int*   A       = (const int*)  d_in[1];   // [3,8192]
  const float* r_count = (const float*)d_in[2];   // [8192]
  const float* w       = (const float*)d_in[3];   // [3,1,512]
  const float* a_sd    = (const float*)d_in[4];   // [4,2,512,1]
  float* out = (float*)d_out;                     // [4,4096,512]

  // Workspace layout (floats), ~35 MB total
  float* ws       = (float*)d_ws;
  float* edge_r   = ws;                                        // E*F
  float* inew     = edge_r + (size_t)N_EDGES * FDIM;           // E*F
  float* T        = inew   + (size_t)N_EDGES * FDIM;           // F*F
  float* svec     = T + (size_t)FDIM * FDIM;                   // F
  float* sc       = svec + FDIM;                               // E
  float* cw       = sc + N_EDGES;                              // 4*F
  float* asrc     = cw + (size_t)N_HEAD * FDIM;                // 4*F
  float* e_rowsum = asrc + (size_t)N_HEAD * FDIM;              // 4*N

  hipMemsetAsync(edge_r, 0, (size_t)N_EDGES * FDIM * sizeof(float), stream);
  hipMemsetAsync(svec,   0, FDIM * sizeof(float), stream);
  hipMemsetAsync(e_rowsum, 0, (size_t)N_HEAD * N_NODES * sizeof(float), stream);
  hipMemsetAsync(out,    0, (size_t)N_HEAD * N_NODES * FDIM * sizeof(float), stream);

  scatter_edge_h_kernel<<<N_EDGES, 256, 0, stream>>>(h, A, r_count, edge_r);
  finalize_edge_r_kernel<<<N_EDGES, 128, 0, stream>>>(edge_r, r_count, svec);
  rowscale_kernel<<<N_EDGES, 256, 0, stream>>>(edge_r, svec, r_count, sc);

  // T = edge_r^T @ diag(sc) @ edge_r   (256 strip-waves)
  gemm_syrk_wmma<<<32, 256, 0, stream>>>(edge_r, sc, T);
  // inputr_new = edge_r @ T            (4096 strip-waves)
  gemm_inew_wmma<<<512, 256, 0, stream>>>(edge_r, T, inew);

  prep_head_kernel<<<2, 256, 0, stream>>>(w, a_sd, cw, asrc);
  head_edge_kernel<<<dim3(N_EDGES, N_HEAD), 256, 0, stream>>>(
      h, A, inew, cw, asrc, a_sd, e_rowsum, out);

  size_t total = (size_t)N_HEAD * N_NODES * FDIM;
  normalize_kernel<<<(unsigned)((total + 255) / 256), 256, 0, stream>>>(out, e_rowsum);
}